// GraphConvModule_80487687127270
// MI455X (gfx1250) — compile-verified
//
#include <hip/hip_runtime.h>

// ---------------------------------------------------------------------------
// 2-layer GCN for MI455X (gfx1250, wave32).
//   out = gcn(relu(gcn(x, W1, b1)), W2, b2) with symmetric normalization and
//   self-loops, exactly mirroring the JAX reference, all in fp32.
//
// Design (see analysis): compute is trivial (1.2 GFLOP); the edge
// gather/scatter phase dominates and its ~90 MB working set lives entirely in
// the 192 MB L2, so plain f32 atomics + coalesced row gathers are the right
// data path. The two dense GEMMs use V_WMMA_F32_16X16X4_F32 so the matrix
// pipeline is exercised with zero precision loss.
// ---------------------------------------------------------------------------

typedef __attribute__((ext_vector_type(2))) float v2f;
typedef __attribute__((ext_vector_type(8))) float v8f;

#define TPB 256

// ---- Stage 1: weighted in-degree (self-loop contributes 1.0) ---------------
__global__ void k_init_deg(float* __restrict__ deg, int n) {
    int i = blockIdx.x * blockDim.x + threadIdx.x;
    if (i < n) deg[i] = 1.0f;  // self-loop weight
}

__global__ void k_edge_deg(const int* __restrict__ dst, const float* __restrict__ ew,
                           float* __restrict__ deg, int E) {
    int e = blockIdx.x * blockDim.x + threadIdx.x;
    if (e < E) atomicAdd(&deg[dst[e]], ew[e]);
}

// deg -> dinv in place: rsqrt(max(deg,1e-12)) (deg>0 guaranteed by self-loop,
// but follow the reference's guard anyway).
__global__ void k_dinv(float* __restrict__ dd, int n) {
    int i = blockIdx.x * blockDim.x + threadIdx.x;
    if (i < n) {
        float d = dd[i];
        dd[i] = (d > 0.0f) ? rsqrtf(fmaxf(d, 1e-12f)) : 0.0f;
    }
}

// ---- Stage 2: dense GEMM H[nrows,NCOL] = A[nrows,K] @ W[K,NCOL] via WMMA ---
// One wave32 per 16x16 output tile; NCOL/16 waves per block handle one
// 16-row stripe. K and NCOL are multiples of 4/16; nrows multiple of 16
// (100000 = 6250*16).
//
// fp32 WMMA operand layouts (CDNA5 ISA 7.12.2):
//   A 16x4 : lane l -> row (l&15), K = 2*(l>>4) + {0,1}   (VGPR0, VGPR1)
//   B 4x16 : lane l -> col (l&15), K = 2*(l>>4) + {0,1}
//   C/D    : lane l, VGPR r -> D[r + 8*(l>>4)][l&15]
template <int K, int NCOL>
__global__ void k_gemm_wmma(const float* __restrict__ A, const float* __restrict__ W,
                            float* __restrict__ H, int nrows) {
    const int lane  = threadIdx.x & 31;
    const int wave  = threadIdx.x >> 5;
    const int row0  = blockIdx.x << 4;
    if (row0 >= nrows) return;
    const int n     = lane & 15;        // A-row index / B-col index
    const int khalf = lane >> 4;        // selects K pair within a K=4 step
    const int ncol0 = wave << 4;

    const float* arow = A + (size_t)(row0 + n) * K + (khalf << 1);
    const float* wptr = W + (size_t)(khalf << 1) * NCOL + ncol0 + n;

    v8f c = {};
#pragma unroll
    for (int kb = 0; kb < K; kb += 4) {
        v2f a = *(const v2f*)(arow + kb);              // A[row][kb+2*khalf+{0,1}]
        v2f b;
        b.x = wptr[(size_t)kb * NCOL];                 // W[kb+2*khalf  ][col]
        b.y = wptr[(size_t)(kb + 1) * NCOL];           // W[kb+2*khalf+1][col]
        c = __builtin_amdgcn_wmma_f32_16x16x4_f32(
                /*neg_a=*/false, a, /*neg_b=*/false, b,
                /*c_mod=*/(short)0, c, /*reuse_a=*/false, /*reuse_b=*/false);
    }

    float* hout = H + (size_t)(row0 + (khalf << 3)) * NCOL + ncol0 + n;
#pragma unroll
    for (int r = 0; r < 8; ++r)
        hout[(size_t)r * NCOL] = c[r];
}

// ---- Stage 3: initialize accumulator with self-loop term + bias ------------
// out[i,f] = dinv[i]^2 * h[i,f] + bias[f]   (self-loop norm = dinv*1*dinv)
// Also serves to fully initialize the (poisoned) accumulation buffer.
template <int F>
__global__ void k_self_bias(const float* __restrict__ h, const float* __restrict__ dinv,
                            const float* __restrict__ bias, float* __restrict__ out, int n) {
    int idx = blockIdx.x * blockDim.x + threadIdx.x;
    if (idx >= n * F) return;
    int i = idx >> (F == 64 ? 6 : 5);
    int f = idx & (F - 1);
    float di = dinv[i];
    out[idx] = di * di * h[idx] + bias[f];
}

// ---- Stage 4: edge scatter. One wave32 per edge. ---------------------------
// F=64: each lane moves 2 floats (coalesced 256B row read -> 64 f32 atomics).
__global__ void k_scatter64(const int* __restrict__ src, const int* __restrict__ dst,
                            const float* __restrict__ ew, const float* __restrict__ dinv,
                            const float* __restrict__ h, float* __restrict__ out, int E) {
    int gid  = blockIdx.x * blockDim.x + threadIdx.x;
    int e    = gid >> 5;
    int lane = threadIdx.x & 31;
    if (e >= E) return;
    int   s = src[e], d = dst[e];
    float w = dinv[s] * ew[e] * dinv[d];
    v2f hv = *(const v2f*)(h + ((size_t)s << 6) + (lane << 1));
    float* op = out + ((size_t)d << 6) + (lane << 1);
    atomicAdd(op, w * hv.x);
    atomicAdd(op + 1, w * hv.y);
}

// F=32: one float per lane.
__global__ void k_scatter32(const int* __restrict__ src, const int* __restrict__ dst,
                            const float* __restrict__ ew, const float* __restrict__ dinv,
                            const float* __restrict__ h, float* __restrict__ out, int E) {
    int gid  = blockIdx.x * blockDim.x + threadIdx.x;
    int e    = gid >> 5;
    int lane = threadIdx.x & 31;
    if (e >= E) return;
    int   s = src[e], d = dst[e];
    float w = dinv[s] * ew[e] * dinv[d];
    float v = h[((size_t)s << 5) + lane];
    atomicAdd(out + ((size_t)d << 5) + lane, w * v);
}

// ---- Stage 5: relu ---------------------------------------------------------
__global__ void k_relu(float* __restrict__ a, int n) {
    int i = blockIdx.x * blockDim.x + threadIdx.x;
    if (i < n) a[i] = fmaxf(a[i], 0.0f);
}

// ---------------------------------------------------------------------------
extern "C" void kernel_launch(void* const* d_in, const int* in_sizes, int n_in,
                              void* d_out, int out_size, void* d_ws, size_t ws_size,
                              hipStream_t stream) {
    (void)n_in; (void)out_size; (void)ws_size;
    const float* x   = (const float*)d_in[0];   // [N, 64]
    const int*   ei  = (const int*)  d_in[1];   // [2, E] (int32: jax default x64-off)
    const float* ew  = (const float*)d_in[2];   // [E]
    const float* W1  = (const float*)d_in[3];   // [64, 64]
    const float* b1  = (const float*)d_in[4];   // [64]
    const float* W2  = (const float*)d_in[5];   // [64, 32]
    const float* b2  = (const float*)d_in[6];   // [32]
    float*       out = (float*)d_out;           // [N, 32]

    const int N = in_sizes[0] / 64;             // 100000 (multiple of 16)
    const int E = in_sizes[2];                  // 1600000
    const int* src = ei;
    const int* dst = ei + E;

    // Workspace layout (all fp32): dinv | h1 (reused as h2) | agg1
    char* ws = (char*)d_ws;
    size_t off = 0;
    float* dinv = (float*)(ws + off);
    off += (((size_t)N * sizeof(float)) + 255) & ~(size_t)255;
    float* h1   = (float*)(ws + off);           // [N,64], later reused for h2 [N,32]
    off += (((size_t)N * 64 * sizeof(float)) + 255) & ~(size_t)255;
    float* agg1 = (float*)(ws + off);           // [N,64]
    float* h2   = h1;

    const int gN     = (N + TPB - 1) / TPB;
    const int gE     = (E + TPB - 1) / TPB;
    const int gEwave = (E + (TPB / 32) - 1) / (TPB / 32);   // one wave per edge
    const int gRows  = (N + 15) / 16;

    // --- normalization ---
    k_init_deg<<<gN, TPB, 0, stream>>>(dinv, N);
    k_edge_deg<<<gE, TPB, 0, stream>>>(dst, ew, dinv, E);
    k_dinv    <<<gN, TPB, 0, stream>>>(dinv, N);

    // --- layer 1: h1 = x @ W1 ; agg1 = scatter(norm * h1[src]) + self + b1 ; relu
    k_gemm_wmma<64, 64><<<gRows, 4 * 32, 0, stream>>>(x, W1, h1, N);
    k_self_bias<64><<<(N * 64 + TPB - 1) / TPB, TPB, 0, stream>>>(h1, dinv, b1, agg1, N);
    k_scatter64<<<gEwave, TPB, 0, stream>>>(src, dst, ew, dinv, h1, agg1, E);
    k_relu<<<(N * 64 + TPB - 1) / TPB, TPB, 0, stream>>>(agg1, N * 64);

    // --- layer 2: h2 = agg1 @ W2 ; out = scatter(norm * h2[src]) + self + b2
    k_gemm_wmma<64, 32><<<gRows, 2 * 32, 0, stream>>>(agg1, W2, h2, N);
    k_self_bias<32><<<(N * 32 + TPB - 1) / TPB, TPB, 0, stream>>>(h2, dinv, b2, out, N);
    k_scatter32<<<gEwave, TPB, 0, stream>>>(src, dst, ew, dinv, h2, out, E);
}